// GCN_63840393887760
// MI455X (gfx1250) — compile-verified
//
#include <hip/hip_runtime.h>
#include <hip/hip_bf16.h>

typedef __attribute__((ext_vector_type(2))) float v2f;
typedef __attribute__((ext_vector_type(8))) float v8f;

#define FEAT 256  // IN_FEAT == OUT_FEAT == 256

// ---------------------------------------------------------------------------
// K0: zero a float4 region (agg matrix + both degree counter arrays)
// ---------------------------------------------------------------------------
__global__ __launch_bounds__(256) void k_zero(float4* __restrict__ p, unsigned n4) {
    unsigned i = blockIdx.x * 256u + threadIdx.x;
    if (i < n4) {
        float4 z; z.x = 0.f; z.y = 0.f; z.z = 0.f; z.w = 0.f;
        p[i] = z;
    }
}

// ---------------------------------------------------------------------------
// K1: degree histograms (u32 atomics -> deterministic counts)
// ---------------------------------------------------------------------------
__global__ __launch_bounds__(256) void k_deg(const int* __restrict__ src,
                                             const int* __restrict__ dst,
                                             unsigned* __restrict__ dego,
                                             unsigned* __restrict__ degi,
                                             int nedges) {
    int i = blockIdx.x * 256 + threadIdx.x;
    if (i < nedges) {
        atomicAdd(&dego[src[i]], 1u);
        atomicAdd(&degi[dst[i]], 1u);
    }
}

// ---------------------------------------------------------------------------
// K2: rinv = rsqrt(max(deg, 1))
// ---------------------------------------------------------------------------
__global__ __launch_bounds__(256) void k_rinv(const unsigned* __restrict__ dego,
                                              const unsigned* __restrict__ degi,
                                              float* __restrict__ ro,
                                              float* __restrict__ ri, int n) {
    int i = blockIdx.x * 256 + threadIdx.x;
    if (i < n) {
        ro[i] = rsqrtf(fmaxf((float)dego[i], 1.0f));
        ri[i] = rsqrtf(fmaxf((float)degi[i], 1.0f));
    }
}

// ---------------------------------------------------------------------------
// K3: transpose W (256x256) -> Wt[col][k], so WMMA B loads are contiguous v2f
// ---------------------------------------------------------------------------
__global__ __launch_bounds__(256) void k_transpose(const float* __restrict__ W,
                                                   float* __restrict__ Wt) {
    int i = blockIdx.x * 256 + threadIdx.x;     // i < 256*256
    int r = i >> 8, c = i & 255;
    Wt[(size_t)c * FEAT + r] = W[(size_t)r * FEAT + c];
}

// ---------------------------------------------------------------------------
// K4: SpMM scatter: agg[dst] += x[src] * rsqrt(deg_out[src])
// 64 threads per edge, float4 per thread, 4 f32 atomics (no-return path)
// ---------------------------------------------------------------------------
__global__ __launch_bounds__(256) void k_spmm(const float4* __restrict__ x4,
                                              const int* __restrict__ src,
                                              const int* __restrict__ dst,
                                              const float* __restrict__ rout,
                                              float* __restrict__ agg,
                                              int nedges) {
    unsigned gid = blockIdx.x * 256u + threadIdx.x;
    unsigned e = gid >> 6;
    if (e >= (unsigned)nedges) return;
    unsigned q = gid & 63u;                     // which float4 of the 256-float row
    int s = src[e], d = dst[e];
    float sc = rout[s];
    float4 v = x4[(size_t)s * 64 + q];
    float* p = agg + (size_t)d * FEAT + q * 4;
    atomicAdd(p + 0, v.x * sc);
    atomicAdd(p + 1, v.y * sc);
    atomicAdd(p + 2, v.z * sc);
    atomicAdd(p + 3, v.w * sc);
}

// ---------------------------------------------------------------------------
// K5: fp32 WMMA GEMM + fused epilogue.
// One wave handles a 16-row tile x all 256 cols.
//   D(16x16) = A(16x4 f32) x B(4x16 f32) + C, K marched 4 at a time.
// A lane layout (ISA 7.12.2): lane l -> row l&15, K = 2*(l>>4) + {vgpr0,vgpr1}
// B symmetric: lane l -> col, K = 2*(l>>4) + {x,y}  (hence the W transpose)
// C/D layout: vgpr r, lanes 0-15 -> row r, lanes 16-31 -> row r+8; col = t*16+(l&15)
// Epilogue: +bias, ReLU, row L2-norm via xor-shuffles within 16-lane halves.
// ---------------------------------------------------------------------------
__global__ __launch_bounds__(256) void k_gemm(const float* __restrict__ agg,
                                              const float* __restrict__ rin,
                                              const float* __restrict__ Wt,
                                              const float* __restrict__ bias,
                                              float* __restrict__ out,
                                              int ntiles, int nrows) {
    int wave = blockIdx.x * (256 / 32) + ((int)threadIdx.x >> 5);
    if (wave >= ntiles) return;                 // wave-uniform: EXEC stays all-1s
    int lane = threadIdx.x & 31;
    int lm = lane & 15;                         // row-in-tile (A) / col-in-tile (B,C)
    int kh = lane >> 4;                         // which K half
    int row0 = wave * 16;

    int rowA = row0 + lm;
    int rowAc = rowA < nrows ? rowA : nrows - 1;   // clamp (branchless, EXEC full)
    float rscale = rin[rowAc];                      // fold right-norm into A

    v8f zero = {};
    v8f acc[16];
#pragma unroll
    for (int t = 0; t < 16; ++t) acc[t] = zero;

    const float* aptr = agg + (size_t)rowAc * FEAT + 2 * kh;

    for (int k = 0; k < FEAT; k += 4) {
        v2f a = *(const v2f*)(aptr + k);
        a.x *= rscale;
        a.y *= rscale;
#pragma unroll
        for (int t = 0; t < 16; ++t) {
            v2f bv = *(const v2f*)(Wt + (size_t)(t * 16 + lm) * FEAT + k + 2 * kh);
            acc[t] = __builtin_amdgcn_wmma_f32_16x16x4_f32(
                /*neg_a=*/false, a, /*neg_b=*/false, bv,
                /*c_mod=*/(short)0, acc[t], /*reuse_a=*/false, /*reuse_b=*/false);
        }
    }

    // bias + ReLU, accumulate per-row sum of squares
    float part[8];
#pragma unroll
    for (int r = 0; r < 8; ++r) part[r] = 0.f;
#pragma unroll
    for (int t = 0; t < 16; ++t) {
        float bv = bias[t * 16 + lm];
#pragma unroll
        for (int r = 0; r < 8; ++r) {
            float v = fmaxf(acc[t][r] + bv, 0.f);
            acc[t][r] = v;
            part[r] += v * v;
        }
    }
    // reduce within each 16-lane half (= within each row of the C layout)
#pragma unroll
    for (int r = 0; r < 8; ++r) {
        float s = part[r];
        s += __shfl_xor(s, 1, 32);
        s += __shfl_xor(s, 2, 32);
        s += __shfl_xor(s, 4, 32);
        s += __shfl_xor(s, 8, 32);
        part[r] = 1.0f / fmaxf(sqrtf(s), 1e-12f);   // F.normalize eps
    }
    // scaled store
#pragma unroll
    for (int t = 0; t < 16; ++t) {
#pragma unroll
        for (int r = 0; r < 8; ++r) {
            int row = row0 + r + 8 * kh;
            if (row < nrows)
                out[(size_t)row * FEAT + t * 16 + lm] = acc[t][r] * part[r];
        }
    }
}

// ---------------------------------------------------------------------------
// Host-side launcher
// ---------------------------------------------------------------------------
static inline size_t align256(size_t x) { return (x + 255) & ~(size_t)255; }

extern "C" void kernel_launch(void* const* d_in, const int* in_sizes, int n_in,
                              void* d_out, int out_size, void* d_ws, size_t ws_size,
                              hipStream_t stream) {
    const float* x   = (const float*)d_in[0];   // [N, 256]
    const float* W   = (const float*)d_in[1];   // [256, 256]
    const float* b   = (const float*)d_in[2];   // [256]
    const int*   src = (const int*)d_in[3];     // [E]
    const int*   dst = (const int*)d_in[4];     // [E]
    float* out = (float*)d_out;

    const int N = in_sizes[0] / FEAT;
    const int E = in_sizes[3];

    // workspace layout
    char* ws = (char*)d_ws;
    size_t off = 0;
    float*    agg  = (float*)(ws + off);    off += align256((size_t)N * FEAT * 4);
    unsigned* dego = (unsigned*)(ws + off); size_t off_dego = off; off += align256((size_t)N * 4);
    unsigned* degi = (unsigned*)(ws + off); off += align256((size_t)N * 4);
    size_t off_deg_end = off;
    float*    ro   = (float*)(ws + off);    off += align256((size_t)N * 4);
    float*    ri   = (float*)(ws + off);    off += align256((size_t)N * 4);
    float*    Wt   = (float*)(ws + off);    off += align256((size_t)FEAT * FEAT * 4);
    (void)off_dego; (void)ws_size;

    // K0: zero agg + both degree arrays in one contiguous float4 sweep
    unsigned n4 = (unsigned)(off_deg_end / 16);
    k_zero<<<(n4 + 255) / 256, 256, 0, stream>>>((float4*)ws, n4);

    // K1: degree histograms
    k_deg<<<(E + 255) / 256, 256, 0, stream>>>(src, dst, dego, degi, E);

    // K2: rsqrt of degrees
    k_rinv<<<(N + 255) / 256, 256, 0, stream>>>(dego, degi, ro, ri, N);

    // K3: transpose W for WMMA-B-friendly loads
    k_transpose<<<(FEAT * FEAT) / 256, 256, 0, stream>>>(W, Wt);

    // K4: SpMM scatter (64 threads per edge)
    {
        unsigned total = (unsigned)E * 64u;
        k_spmm<<<(total + 255) / 256, 256, 0, stream>>>((const float4*)x, src, dst,
                                                        ro, agg, E);
    }

    // K5: WMMA GEMM + bias + ReLU + L2 normalize
    {
        int ntiles = (N + 15) / 16;
        int waves_per_block = 256 / 32;
        int blocks = (ntiles + waves_per_block - 1) / waves_per_block;
        k_gemm<<<blocks, 256, 0, stream>>>(agg, ri, Wt, b, out, ntiles, N);
    }
}